// TripletRankingLoss_58299886076038
// MI455X (gfx1250) — compile-verified
//
#include <hip/hip_runtime.h>
#include <stdint.h>

#define BROWS 8192
#define DIMS 512
#define HALF_N (1u << 25)   // (B*B)/2 : threefry counter offset between paired rows
#define MARGINV 0.1f
#define EPSV 1e-6f

typedef float v2f __attribute__((ext_vector_type(2)));
typedef float v8f __attribute__((ext_vector_type(8)));

// ---------------- exact JAX threefry2x32 ----------------
struct U2 { uint32_t x, y; };

__host__ __device__ constexpr uint32_t rotl32(uint32_t v, int r) {
    return (v << r) | (v >> (32 - r));
}

__host__ __device__ constexpr U2 threefry2x32(uint32_t k0, uint32_t k1,
                                              uint32_t c0, uint32_t c1) {
    uint32_t ks2 = k0 ^ k1 ^ 0x1BD11BDAu;
    uint32_t x0 = c0 + k0, x1 = c1 + k1;
#define TF_R(r) { x0 += x1; x1 = rotl32(x1, r); x1 ^= x0; }
    TF_R(13) TF_R(15) TF_R(26) TF_R(6)
    x0 += k1;  x1 += ks2 + 1u;
    TF_R(17) TF_R(29) TF_R(16) TF_R(24)
    x0 += ks2; x1 += k0 + 2u;
    TF_R(13) TF_R(15) TF_R(26) TF_R(6)
    x0 += k0;  x1 += k1 + 3u;
    TF_R(17) TF_R(29) TF_R(16) TF_R(24)
    x0 += k1;  x1 += ks2 + 4u;
    TF_R(13) TF_R(15) TF_R(26) TF_R(6)
    x0 += ks2; x1 += k0 + 5u;
#undef TF_R
    return U2{x0, x1};
}

// jax.random.key(42) -> (0,42); split(key) -> kp, kn (compile-time constants)
constexpr U2 SPLIT_A = threefry2x32(0u, 42u, 0u, 2u);
constexpr U2 SPLIT_B = threefry2x32(0u, 42u, 1u, 3u);
constexpr uint32_t KP0 = SPLIT_A.x, KP1 = SPLIT_B.x;  // g_pos key
constexpr uint32_t KN0 = SPLIT_A.y, KN1 = SPLIT_B.y;  // g_neg key

// ---------------- kernel 0: zero the output accumulator ----------------
__global__ void zero_out_kernel(float* out) { out[0] = 0.0f; }

// ---------------- kernel 1: row norms, clamped at eps ----------------
__global__ __launch_bounds__(256) void norms_kernel(const float* __restrict__ pred,
                                                    float* __restrict__ norms) {
    int wave = (int)((blockIdx.x * blockDim.x + threadIdx.x) >> 5);
    int lane = (int)(threadIdx.x & 31);
    if (wave >= BROWS) return;
    const float4* row4 = (const float4*)(pred + (size_t)wave * DIMS);
    float s = 0.0f;
#pragma unroll
    for (int it = 0; it < DIMS / (32 * 4); ++it) {
        float4 v = row4[lane + it * 32];
        s += v.x * v.x + v.y * v.y + v.z * v.z + v.w * v.w;
    }
#pragma unroll
    for (int off = 16; off; off >>= 1) s += __shfl_xor(s, off);
    if (lane == 0) norms[wave] = fmaxf(sqrtf(s), EPSV);
}

// ---------------- kernel 2: gumbel-argmax sampling of pos/neg indices ----------
// One wave per row-pair (i, i+4096): threefry(key,(i*8192+j, +2^25)) yields the
// random bits for BOTH rows at column j. Monotonicity: argmax over gumbel ==
// argmax over (bits>>9); ties -> smallest j (matches jnp.argmax).
__global__ __launch_bounds__(256) void sample_kernel(const int* __restrict__ labels,
                                                     uint32_t* __restrict__ idx_p,
                                                     uint32_t* __restrict__ idx_n) {
    __shared__ int lbl[BROWS];
    for (int t = (int)threadIdx.x; t < BROWS; t += (int)blockDim.x) lbl[t] = labels[t];
    __syncthreads();

    int wave = (int)(threadIdx.x >> 5);
    int lane = (int)(threadIdx.x & 31);
    int i  = (int)blockIdx.x * 8 + wave;   // [0, 4096)
    int i2 = i + (BROWS / 2);
    int Li = lbl[i], Li2 = lbl[i2];
    uint32_t base0 = (uint32_t)i * (uint32_t)BROWS;

    unsigned long long bpA = 0ull, bnA = 0ull, bpB = 0ull, bnB = 0ull;

    for (int j = lane; j < BROWS; j += 32) {
        uint32_t c0 = base0 + (uint32_t)j;
        uint32_t c1 = c0 + HALF_N;
        U2 gp = threefry2x32(KP0, KP1, c0, c1);
        U2 gn = threefry2x32(KN0, KN1, c0, c1);
        int lj = lbl[j];
        unsigned long long jc = (unsigned long long)(uint32_t)(BROWS - 1 - j);
        unsigned long long kpA = (((unsigned long long)((gp.x >> 9) + 1u)) << 32) | jc;
        unsigned long long kpB = (((unsigned long long)((gp.y >> 9) + 1u)) << 32) | jc;
        unsigned long long knA = (((unsigned long long)((gn.x >> 9) + 1u)) << 32) | jc;
        unsigned long long knB = (((unsigned long long)((gn.y >> 9) + 1u)) << 32) | jc;
        if (lj == Li)  { if (kpA > bpA) bpA = kpA; } else { if (knA > bnA) bnA = knA; }
        if (lj == Li2) { if (kpB > bpB) bpB = kpB; } else { if (knB > bnB) bnB = knB; }
    }
#pragma unroll
    for (int off = 16; off; off >>= 1) {
        unsigned long long t;
        t = __shfl_xor(bpA, off); if (t > bpA) bpA = t;
        t = __shfl_xor(bnA, off); if (t > bnA) bnA = t;
        t = __shfl_xor(bpB, off); if (t > bpB) bpB = t;
        t = __shfl_xor(bnB, off); if (t > bnB) bnB = t;
    }
    if (lane == 0) {
        idx_p[i]  = (uint32_t)(BROWS - 1) - (uint32_t)bpA;
        idx_n[i]  = (uint32_t)(BROWS - 1) - (uint32_t)bnA;
        idx_p[i2] = (uint32_t)(BROWS - 1) - (uint32_t)bpB;
        idx_n[i2] = (uint32_t)(BROWS - 1) - (uint32_t)bnB;
    }
}

// ---------------- kernel 3: WMMA cosine dots + hinge + mean --------------------
// One wave per 16-anchor tile. Full f32 WMMA (16x16x4) accumulates A*P^T and
// A*N^T over K=512; only the diagonal is consumed.
// A-frag layout (ISA 16x4 f32): lane l<16 holds A[l][k..k+1]; lane>=16 holds
// A[l-16][k+2..k+3]. B-frag (4x16) mirrors it with B[kk][n] = P[n][k+kk], so
// both fragments are the same per-lane float2 load from row (lane&15).
__global__ __launch_bounds__(256) void loss_kernel(const float* __restrict__ pred,
                                                   const float* __restrict__ norms,
                                                   const uint32_t* __restrict__ idx_p,
                                                   const uint32_t* __restrict__ idx_n,
                                                   float* __restrict__ out) {
    int wave = (int)(threadIdx.x >> 5);
    int lane = (int)(threadIdx.x & 31);
    int tile = (int)blockIdx.x * 8 + wave;        // [0, 512)
    int m = lane & 15;
    int halfsel = lane >> 4;
    int rowA = tile * 16 + m;
    uint32_t rowP = idx_p[rowA];
    uint32_t rowN = idx_n[rowA];
    const float* pa = pred + (size_t)rowA * DIMS;
    const float* pp = pred + (size_t)rowP * DIMS;
    const float* pn = pred + (size_t)rowN * DIMS;
    int cbase = halfsel * 2;

    v8f accP = {};
    v8f accN = {};
    for (int k = 0; k < DIMS; k += 4) {
        v2f a = *(const v2f*)(pa + k + cbase);
        v2f p = *(const v2f*)(pp + k + cbase);
        v2f n = *(const v2f*)(pn + k + cbase);
        accP = __builtin_amdgcn_wmma_f32_16x16x4_f32(false, a, false, p,
                                                     (short)0, accP, false, false);
        accN = __builtin_amdgcn_wmma_f32_16x16x4_f32(false, a, false, n,
                                                     (short)0, accN, false, false);
    }

    // diagonal element M=N=mm lives in VGPR (mm&7) at lane mm (mm<8) or mm+16 (mm>=8)
    float term = 0.0f;
#pragma unroll
    for (int r = 0; r < 8; ++r) {
        int mm = -1;
        if (lane == r) mm = r;
        else if (lane == r + 24) mm = r + 8;
        if (mm >= 0) {
            int gr = tile * 16 + mm;
            float nA = norms[gr];
            float nP = norms[idx_p[gr]];
            float nN = norms[idx_n[gr]];
            float c = accP[r] / (nA * nP) - accN[r] / (nA * nN) + MARGINV;
            term = fmaxf(c, 0.0f);
        }
    }
#pragma unroll
    for (int off = 16; off; off >>= 1) term += __shfl_xor(term, off);
    if (lane == 0) atomicAdd(out, term * (1.0f / (float)BROWS));
}

extern "C" void kernel_launch(void* const* d_in, const int* in_sizes, int n_in,
                              void* d_out, int out_size, void* d_ws, size_t ws_size,
                              hipStream_t stream) {
    (void)in_sizes; (void)n_in; (void)out_size; (void)ws_size;
    const float* pred  = (const float*)d_in[0];
    const int* labels  = (const int*)d_in[1];
    float* out = (float*)d_out;

    float*    norms = (float*)d_ws;
    uint32_t* idx_p = (uint32_t*)((char*)d_ws + BROWS * sizeof(float));
    uint32_t* idx_n = idx_p + BROWS;

    zero_out_kernel<<<1, 1, 0, stream>>>(out);
    norms_kernel<<<BROWS / 8, 256, 0, stream>>>(pred, norms);
    sample_kernel<<<(BROWS / 2) / 8, 256, 0, stream>>>(labels, idx_p, idx_n);
    loss_kernel<<<(BROWS / 16) / 8, 256, 0, stream>>>(pred, norms, idx_p, idx_n, out);
}